// DeepseekV3TopkRouter_11613591568696
// MI455X (gfx1250) — compile-verified
//
#include <hip/hip_runtime.h>

// ---------------- problem constants (from reference) ----------------
#define TOP_K        8
#define N_EXPERTS    256
#define N_GROUP      8
#define TOPK_GROUP   4
#define PER_GROUP    32           // N_EXPERTS / N_GROUP
#define HIDDEN       7168
#define TOKENS       16384
#define ROUTED_SCALING 2.5f

// ---------------- tiling ----------------
#define TOK_TILE     16           // tokens per block (one WMMA M-tile)
#define KC           64           // K chunk staged in LDS
#define THREADS      512          // 16 waves (wave32): one 16-expert tile per wave

typedef __attribute__((ext_vector_type(2))) float v2f;
typedef __attribute__((ext_vector_type(8))) float v8f;

__global__ __launch_bounds__(THREADS)
void DeepseekV3TopkRouter_kernel(const float* __restrict__ hidden,
                                 const float* __restrict__ weight,
                                 const float* __restrict__ bias,
                                 int*   __restrict__ out_idx,
                                 float* __restrict__ out_w)
{
    // A slab: 16 tokens x KC, +4 pad floats/row -> per-lane float2 fragment
    // reads land on 64 distinct banks (stride 68 floats).
    __shared__ float sA[TOK_TILE][KC + 4];
    // Scores: 16 tokens x 256 experts, +1 pad to kill half-wave write conflicts.
    __shared__ float sScore[TOK_TILE][N_EXPERTS + 1];
    __shared__ float sBias[N_EXPERTS];

    const int tid     = threadIdx.x;
    const int lane    = tid & 31;
    const int wave    = tid >> 5;
    const int tokBase = blockIdx.x * TOK_TILE;
    const int expBase = wave * 16;                 // 16 experts per wave tile
    const int nrow    = lane & 15;                 // M (for A) / N (for B) index
    const int halfSel = (lane < 16) ? 0 : 2;       // K sub-offset per half-wave

    if (tid < N_EXPERTS) sBias[tid] = bias[tid];

    // Per-lane B row pointer: B[k][n] = weight[expBase + n][k]
    const float* wrow = weight + (size_t)(expBase + nrow) * HIDDEN;

    v8f acc = {0.f, 0.f, 0.f, 0.f, 0.f, 0.f, 0.f, 0.f};

    for (int k0 = 0; k0 < HIDDEN; k0 += KC) {
        __syncthreads();   // protect sA readers from previous chunk
        {
            // 512 threads cooperatively stage 16 x KC floats (one float2 each),
            // fully coalesced: each wave covers one 64-float row chunk.
            const int r = tid >> 5;
            const int c = (tid & 31) << 1;
            const float2 v =
                *(const float2*)(hidden + (size_t)(tokBase + r) * HIDDEN + k0 + c);
            *(float2*)&sA[r][c] = v;
        }
        __syncthreads();

        #pragma unroll
        for (int kk = 0; kk < KC; kk += 4) {
            // A 16x4 fragment: lane<16 -> {A[m][k],A[m][k+1]}, lane>=16 -> {A[m][k+2],A[m][k+3]}
            v2f a = *(const v2f*)&sA[nrow][kk + halfSel];
            // B 4x16 fragment mirrors A with N striped across lanes; contiguous
            // float2 from the weight row (L2-resident, 7.3 MB total).
            v2f b = *(const v2f*)(wrow + k0 + kk + halfSel);
            // D = A x B + C  (fp32, 16x16x4)
            acc = __builtin_amdgcn_wmma_f32_16x16x4_f32(
                      /*neg_a=*/false, a, /*neg_b=*/false, b,
                      /*c_mod=*/(short)0, acc,
                      /*reuse_a=*/false, /*reuse_b=*/false);
        }
    }

    // Scatter C tile into LDS: row = i + (lane>=16 ? 8 : 0), col = expBase + (lane&15)
    {
        const int rofs = (lane >= 16) ? 8 : 0;
        #pragma unroll
        for (int i = 0; i < 8; ++i)
            sScore[i + rofs][expBase + nrow] = acc[i];
    }
    __syncthreads();

    // sigmoid + bias, in place (corrected scores); raw = corrected - bias later.
    for (int i = tid; i < TOK_TILE * N_EXPERTS; i += THREADS) {
        const int r = i >> 8;
        const int e = i & 255;
        const float x = sScore[r][e];
        const float s = 1.0f / (1.0f + expf(-x));
        sScore[r][e] = s + sBias[e];
    }
    __syncthreads();

    // ---------------- routing: one thread per token ----------------
    if (tid < TOK_TILE) {
        float* sc = sScore[tid];

        // group score = sum of top-2 corrected scores in each group of 32
        float gs[N_GROUP];
        #pragma unroll
        for (int g = 0; g < N_GROUP; ++g) {
            float m1 = -3.4e38f, m2 = -3.4e38f;
            const float* p = sc + g * PER_GROUP;
            for (int j = 0; j < PER_GROUP; ++j) {
                const float v = p[j];
                if (v > m1)      { m2 = m1; m1 = v; }
                else if (v > m2) { m2 = v; }
            }
            gs[g] = m1 + m2;
        }

        // top-4 groups (ties -> lowest group index, matching jax top_k)
        unsigned gsel = 0u;
        for (int it = 0; it < TOPK_GROUP; ++it) {
            float best = -3.4e38f; int bg = 0;
            for (int g = 0; g < N_GROUP; ++g)
                if (!((gsel >> g) & 1u) && gs[g] > best) { best = gs[g]; bg = g; }
            gsel |= (1u << bg);
        }

        // top-8 experts over where(group_selected, corrected, 0.0)
        unsigned used[8] = {0, 0, 0, 0, 0, 0, 0, 0};
        int   idxs[TOP_K];
        float wts[TOP_K];
        float wsum = 0.0f;
        for (int it = 0; it < TOP_K; ++it) {
            float best = -3.4e38f; int bi = 0;
            for (int e = 0; e < N_EXPERTS; ++e) {
                const int g = e >> 5;
                const bool isUsed = (used[g] >> (e & 31)) & 1u;
                const float v = isUsed ? -3.4e38f
                                       : (((gsel >> g) & 1u) ? sc[e] : 0.0f);
                if (v > best) { best = v; bi = e; }
            }
            used[bi >> 5] |= (1u << (bi & 31));
            idxs[it] = bi;
            const float raw = sc[bi] - sBias[bi];   // raw sigmoid score
            wts[it] = raw;
            wsum   += raw;
        }

        const float scale = ROUTED_SCALING / (wsum + 1e-20f);
        const int tok = tokBase + tid;
        #pragma unroll
        for (int k = 0; k < TOP_K; ++k) {
            out_idx[tok * TOP_K + k] = idxs[k];
            out_w  [tok * TOP_K + k] = wts[k] * scale;
        }
    }
}

extern "C" void kernel_launch(void* const* d_in, const int* in_sizes, int n_in,
                              void* d_out, int out_size, void* d_ws, size_t ws_size,
                              hipStream_t stream) {
    (void)in_sizes; (void)n_in; (void)out_size; (void)d_ws; (void)ws_size;
    const float* hidden = (const float*)d_in[0];   // [16384, 7168] f32
    const float* weight = (const float*)d_in[1];   // [256, 7168]   f32
    const float* bias   = (const float*)d_in[2];   // [256]         f32

    // d_out = [topk_indices (int32, 16384*8)] ++ [topk_weights (f32, 16384*8)]
    int*   out_idx = (int*)d_out;
    float* out_w   = (float*)d_out + (size_t)TOKENS * TOP_K;

    dim3 grid(TOKENS / TOK_TILE);   // 1024 blocks
    dim3 block(THREADS);            // 16 waves
    hipLaunchKernelGGL(DeepseekV3TopkRouter_kernel, grid, block, 0, stream,
                       hidden, weight, bias, out_idx, out_w);
}